// EnergyBiasModule_83545703842024
// MI455X (gfx1250) — compile-verified
//
#include <hip/hip_runtime.h>
#include <math.h>
#include <stdint.h>

#define B_ 32
#define T_ 4096
#define H_ 100
#define M_ 65
#define DELAY_ 150
#define TSHIFT 12  // log2(T_)

// f32(2*pi) exactly as the reference's weakly-typed f32 math uses it
#define TWO_PI_F 6.2831853071795864769f
// offsets = linspace(0, 1.5*pi, 100): step
#define OFF_STEP_F ((float)(4.712388980384690 / 99.0))
// ripple_base step = 2.5*pi
#define RB_STEP_F 7.853981633974483f
// phase_inc = 2*pi*(25+0.5*30)*64/16000
#define PH_INC_F ((float)(6.283185307179586 * 40.0 * 64.0 / 16000.0))

#ifndef __has_builtin
#define __has_builtin(x) 0
#endif

#if __has_builtin(__builtin_amdgcn_global_load_async_to_lds_b128)
#define ASYNC_MODE 2
#elif __has_builtin(__builtin_amdgcn_global_load_async_to_lds_b32)
#define ASYNC_MODE 1
#else
#define ASYNC_MODE 0
#endif

typedef int v4i __attribute__((ext_vector_type(4)));

#if ASYNC_MODE
__device__ __forceinline__ void wait_async_zero() {
#if __has_builtin(__builtin_amdgcn_s_wait_asynccnt)
  __builtin_amdgcn_s_wait_asynccnt(0);
#else
  asm volatile("s_wait_asynccnt 0" ::: "memory");
#endif
}
#endif

// ---- wave32 reductions ------------------------------------------------------
__device__ __forceinline__ float wred_max(float v) {
#pragma unroll
  for (int s = 16; s > 0; s >>= 1) v = fmaxf(v, __shfl_xor(v, s, 32));
  return v;
}
__device__ __forceinline__ float wred_sum(float v) {
#pragma unroll
  for (int s = 16; s > 0; s >>= 1) v += __shfl_xor(v, s, 32);
  return v;
}

// phase[t] of the reference's sequential fmod chain (exact arithmetic in f64,
// modulus = f32(2*pi) to match the reference's f32 '%')
__device__ __forceinline__ float phase_at(int t) {
  return (float)fmod((double)(t + 1) * (double)PH_INC_F, (double)TWO_PI_F);
}
// x^3.4 via raw v_log_f32 / v_exp_f32 (x clamped >= 1e-6 by caller)
__device__ __forceinline__ float pow_g(float x) {
  return __builtin_amdgcn_exp2f(3.4f * __builtin_amdgcn_logf(x));
}
__device__ __forceinline__ float fsin(float x) { return __builtin_amdgcn_sinf(x); }

// ============================================================================
// K1: per-frame t — tension + turbulence, batch-mean centroid c_mean[t] and
//     batch-mean turbulent spectrum meanturb[t][i]. One block per t; rows of
//     all 32 batches staged into LDS via async-to-LDS b128.
// ============================================================================
__global__ void __launch_bounds__(128) k1_centroid_meanturb(
    const float* __restrict__ harm, float* __restrict__ c_mean,
    float* __restrict__ meanturb) {
  const int t = blockIdx.x;
  const int tid = threadIdx.x;
  const int w = tid >> 5;
  const int lane = tid & 31;

  __shared__ __align__(16) float rowsf[B_ * H_];  // 12.8 KB
  __shared__ float tile[4][H_];
  __shared__ float centw[4];

#if ASYNC_MODE == 2
  for (int q = tid; q < (B_ * H_) / 4; q += 128) {
    const int e = q << 2;
    const int b = e / H_;
    const int i = e - b * H_;
    const float* g = harm + ((size_t)b * T_ + t) * H_ + i;
    __builtin_amdgcn_global_load_async_to_lds_b128((v4i*)(void*)g,
                                                   (v4i*)(void*)&rowsf[e], 0, 0);
  }
  wait_async_zero();
#elif ASYNC_MODE == 1
  for (int e = tid; e < B_ * H_; e += 128) {
    const int b = e / H_;
    const int i = e - b * H_;
    const float* g = harm + ((size_t)b * T_ + t) * H_ + i;
    __builtin_amdgcn_global_load_async_to_lds_b32((int*)(void*)g,
                                                  (int*)(void*)&rowsf[e], 0, 0);
  }
  wait_async_zero();
#else
  for (int q = tid; q < (B_ * H_) / 4; q += 128) {
    const int e = q << 2;
    const int b = e / H_;
    const int i = e - b * H_;
    *(float4*)&rowsf[e] = *(const float4*)(harm + ((size_t)b * T_ + t) * H_ + i);
  }
#endif
  __syncthreads();

  const float phase_t = phase_at(t);
  const bool act = (lane < (H_ >> 2));  // 25 lanes x float4 = 100 elems
  const int i0 = lane << 2;
  float a0 = 0.f, a1 = 0.f, a2 = 0.f, a3 = 0.f;
  float csum = 0.f;

  for (int b = w; b < B_; b += 4) {  // 8 rows per wave
    float4 v = make_float4(0.f, 0.f, 0.f, 0.f);
    if (act) v = *(const float4*)&rowsf[b * H_ + i0];
    const float s0 = pow_g(fmaxf(v.x, 1e-6f));
    const float s1 = pow_g(fmaxf(v.y, 1e-6f));
    const float s2 = pow_g(fmaxf(v.z, 1e-6f));
    const float s3 = pow_g(fmaxf(v.w, 1e-6f));
    const float mo = wred_max(fmaxf(fmaxf(v.x, v.y), fmaxf(v.z, v.w)));
    const float ms = wred_max(fmaxf(fmaxf(s0, s1), fmaxf(s2, s3)));
    const float scale = fmaxf(mo, 1e-6f) / fmaxf(ms, 1e-6f);
    float hs = 0.f, his = 0.f;
    if (act) {
      const float h0 = s0 * scale * (1.f + 0.225f * fsin(phase_t + (float)(i0 + 0) * OFF_STEP_F));
      const float h1 = s1 * scale * (1.f + 0.225f * fsin(phase_t + (float)(i0 + 1) * OFF_STEP_F));
      const float h2 = s2 * scale * (1.f + 0.225f * fsin(phase_t + (float)(i0 + 2) * OFF_STEP_F));
      const float h3 = s3 * scale * (1.f + 0.225f * fsin(phase_t + (float)(i0 + 3) * OFF_STEP_F));
      a0 += h0; a1 += h1; a2 += h2; a3 += h3;
      hs = h0 + h1 + h2 + h3;
      his = h0 * (float)(i0) + h1 * (float)(i0 + 1) + h2 * (float)(i0 + 2) + h3 * (float)(i0 + 3);
    }
    hs = wred_sum(hs);
    his = wred_sum(his);
    csum += his / fmaxf(hs, 1e-6f);
  }
  if (act) {
    tile[w][i0 + 0] = a0; tile[w][i0 + 1] = a1;
    tile[w][i0 + 2] = a2; tile[w][i0 + 3] = a3;
  }
  if (lane == 0) centw[w] = csum;
  __syncthreads();
  if (tid < H_) {
    meanturb[(size_t)t * H_ + tid] =
        (tile[0][tid] + tile[1][tid] + tile[2][tid] + tile[3][tid]) * (1.f / (float)B_);
  }
  if (tid == 0) {
    c_mean[t] = (centw[0] + centw[1] + centw[2] + centw[3]) * (1.f / (float)B_);
  }
}

// ============================================================================
// K2: fb affine recurrence + clock prefix, parallelized as a wave32 affine-map
//     scan in f64 (lane = 128-frame chunk). Emits rrs[t] = 0.4*sin(clock[t]).
// ============================================================================
__global__ void __launch_bounds__(32) k2_fb_clock(const float* __restrict__ c_mean,
                                                  float* __restrict__ rrs) {
  const int lane = threadIdx.x;
  const int CH = T_ / 32;  // 128
  const int base = lane * CH;
  const double TWO_PI_D32 = (double)TWO_PI_F;  // reference mods by f32(2*pi)
  const double FRAME_D = (double)0.004f;       // f32(64/16000)

  // pass 1: chunk-local fb with fb_in = 0; chunk map is x -> A*x + p
  double fb = 0.0, A = 1.0;
  for (int k = 0; k < CH; ++k) {
    const double g = ((double)c_mean[base + k] - 30.0) / 40.0;
    fb = 0.9 * fb + 0.1 * g;
    A *= 0.9;
  }
  double a = A, p = fb;
#pragma unroll
  for (int s = 1; s < 32; s <<= 1) {  // inclusive scan of affine maps
    const double au = __shfl_up(a, s, 32);
    const double pu = __shfl_up(p, s, 32);
    if (lane >= s) { p = a * pu + p; a = a * au; }
  }
  double fb_in = __shfl_up(p, 1, 32);
  if (lane == 0) fb_in = 0.0;

  // pass 2: chunk rate totals for the clock prefix
  double fb2 = fb_in, rsum = 0.0;
  for (int k = 0; k < CH; ++k) {
    const double g = ((double)c_mean[base + k] - 30.0) / 40.0;
    fb2 = 0.9 * fb2 + 0.1 * g;
    rsum += ((0.725 * (1.0 + 0.4 * fb2)) * TWO_PI_D32) * FRAME_D;
  }
  double run = rsum;
#pragma unroll
  for (int s = 1; s < 32; s <<= 1) {
    const double u = __shfl_up(run, s, 32);
    if (lane >= s) run += u;
  }
  double clock = run - rsum;  // exclusive prefix = chunk's clock base

  // pass 3: emit rrs
  fb2 = fb_in;
  for (int k = 0; k < CH; ++k) {
    const double g = ((double)c_mean[base + k] - 30.0) / 40.0;
    fb2 = 0.9 * fb2 + 0.1 * g;
    clock += ((0.725 * (1.0 + 0.4 * fb2)) * TWO_PI_D32) * FRAME_D;
    rrs[base + k] = 0.4f * sinf((float)fmod(clock, TWO_PI_D32));
  }
}

// ============================================================================
// K3: recompute tension+turbulence, apply resonance, blend the 149-frame
//     delayed batch-mean memory, write harmonic output. One wave per row.
// ============================================================================
__global__ void __launch_bounds__(256) k3_harm_out(
    const float* __restrict__ harm, const float* __restrict__ rrs,
    const float* __restrict__ meanturb, float* __restrict__ out_h) {
  const int wid = (blockIdx.x << 3) + (threadIdx.x >> 5);
  const int lane = threadIdx.x & 31;
  const int b = wid >> TSHIFT;
  const int t = wid & (T_ - 1);
  if (b >= B_) return;
  const bool act = (lane < (H_ >> 2));
  const int i0 = lane << 2;
  const size_t row = ((size_t)b * T_ + t) * H_;
  const float phase_t = phase_at(t);
  const float rr = rrs[t];
  const bool has_mem = (t >= DELAY_ - 1);  // read slot written at t-149
  const float rrp = has_mem ? rrs[t - (DELAY_ - 1)] : 0.f;

  float4 v = make_float4(0.f, 0.f, 0.f, 0.f);
  if (act) v = *(const float4*)(harm + row + i0);
  const float s0 = pow_g(fmaxf(v.x, 1e-6f));
  const float s1 = pow_g(fmaxf(v.y, 1e-6f));
  const float s2 = pow_g(fmaxf(v.z, 1e-6f));
  const float s3 = pow_g(fmaxf(v.w, 1e-6f));
  const float mo = wred_max(fmaxf(fmaxf(v.x, v.y), fmaxf(v.z, v.w)));
  const float ms = wred_max(fmaxf(fmaxf(s0, s1), fmaxf(s2, s3)));
  const float scale = fmaxf(mo, 1e-6f) / fmaxf(ms, 1e-6f);
  if (!act) return;  // no shuffles/barriers past this point

  float4 mv = make_float4(0.f, 0.f, 0.f, 0.f);
  if (has_mem) mv = *(const float4*)(meanturb + (size_t)(t - (DELAY_ - 1)) * H_ + i0);

  auto elem = [&](float s, float mval, int i) -> float {
    const float hr = (float)i / 99.0f * 2.0f - 1.0f;
    const float ht = s * scale * (1.f + 0.225f * fsin(phase_t + (float)i * OFF_STEP_F));
    const float hres = ht * (1.f + rr * hr);
    if (t == 0) return hres;  // first frame passes through
    const float m = has_mem ? mval * (1.f + rrp * hr) : 0.f;
    return 0.6f * hres + 0.4f * m;
  };
  float4 o;
  o.x = elem(s0, mv.x, i0 + 0);
  o.y = elem(s1, mv.y, i0 + 1);
  o.z = elem(s2, mv.z, i0 + 2);
  o.w = elem(s3, mv.w, i0 + 3);
  *(float4*)(out_h + row + i0) = o;
}

// ============================================================================
// K4: noise ripple AM (embarrassingly parallel)
// ============================================================================
__global__ void __launch_bounds__(256) k4_noise_out(const float* __restrict__ nm,
                                                    float* __restrict__ out_n) {
  const unsigned idx = blockIdx.x * 256u + threadIdx.x;
  if (idx >= (unsigned)(B_ * T_ * M_)) return;
  const unsigned m = idx % (unsigned)M_;
  const unsigned t = (idx / (unsigned)M_) & (unsigned)(T_ - 1);
  const float rb = (float)m * RB_STEP_F;
  out_n[idx] = nm[idx] * (1.0f + 0.6f * fsin(rb + phase_at((int)t)));
}

// ============================================================================
extern "C" void kernel_launch(void* const* d_in, const int* in_sizes, int n_in,
                              void* d_out, int out_size, void* d_ws, size_t ws_size,
                              hipStream_t stream) {
  (void)in_sizes; (void)n_in; (void)out_size;
  const float* harm = (const float*)d_in[0];
  const float* noise = (const float*)d_in[1];
  float* out_h = (float*)d_out;
  float* out_n = out_h + (size_t)B_ * T_ * H_;

  // scratch: c_mean[T] | rrs[T] | meanturb[T*H]  (~1.67 MB)
  const size_t need = ((size_t)2 * T_ + (size_t)T_ * H_) * sizeof(float);
  // fall back to the noise-output region (overwritten by K4 afterwards) if
  // the provided workspace is too small — deterministic either way.
  float* scratch = (ws_size >= need) ? (float*)d_ws : out_n;
  float* c_mean = scratch;
  float* rrs = scratch + T_;
  float* meanturb = scratch + 2 * T_;

  k1_centroid_meanturb<<<T_, 128, 0, stream>>>(harm, c_mean, meanturb);
  k2_fb_clock<<<1, 32, 0, stream>>>(c_mean, rrs);
  k3_harm_out<<<(B_ * T_) / 8, 256, 0, stream>>>(harm, rrs, meanturb, out_h);
  k4_noise_out<<<((unsigned)(B_ * T_ * M_) + 255u) / 256u, 256, 0, stream>>>(noise, out_n);
}